// VariationalGCNEncoder_23441931501600
// MI455X (gfx1250) — compile-verified
//
#include <hip/hip_runtime.h>

typedef __attribute__((ext_vector_type(2))) float v2f;
typedef __attribute__((ext_vector_type(8))) float v8f;

#define C_IN 128
#define HID 64
#define C_OUT 32

// ---------------- elementwise / init kernels ----------------

__global__ void k_set1(float* __restrict__ p, int n) {
  int i = blockIdx.x * blockDim.x + threadIdx.x;
  if (i < n) p[i] = 1.0f;  // self-loop contributes 1 to every degree
}

__global__ void k_zero(float* __restrict__ p, int n) {
  int i = blockIdx.x * blockDim.x + threadIdx.x;
  if (i < n) p[i] = 0.0f;
}

__global__ void k_deg_acc(const long long* __restrict__ dst,
                          float* __restrict__ deg, int E) {
  int e = blockIdx.x * blockDim.x + threadIdx.x;
  if (e < E) atomicAdd(&deg[(int)dst[e]], 1.0f);
}

__global__ void k_rsqrt(float* __restrict__ p, int n) {
  int i = blockIdx.x * blockDim.x + threadIdx.x;
  if (i < n) {
    float d = p[i];
    p[i] = d > 0.0f ? rsqrtf(d) : 0.0f;
  }
}

// Wt[n*C_IN + k] = W1[k*HID + n]   (transpose 128x64 -> 64x128)
__global__ void k_transpose_w1(const float* __restrict__ W, float* __restrict__ Wt) {
  int idx = blockIdx.x * blockDim.x + threadIdx.x;
  if (idx < C_IN * HID) {
    int k = idx / HID, n = idx % HID;
    Wt[n * C_IN + k] = W[idx];
  }
}

// Wt[n*HID + k] = [W_mu | W_ls][k, n]  (transposed concat, 64x64)
__global__ void k_build_wcat(const float* __restrict__ Wmu,
                             const float* __restrict__ Wls,
                             float* __restrict__ Wt) {
  int idx = blockIdx.x * blockDim.x + threadIdx.x;
  if (idx < HID * HID) {
    int k = idx / HID, n = idx % HID;
    float v = (n < C_OUT) ? Wmu[k * C_OUT + n] : Wls[k * C_OUT + (n - C_OUT)];
    Wt[n * HID + k] = v;
  }
}

// ---------------- WMMA fp32 GEMM: C[M,64] = A[M,K] * B[K,64] ----------------
// Bt is B pre-transposed: Bt[n*K + k]. One wave computes a 16x64 output tile
// via 4 accumulator fragments of V_WMMA_F32_16X16X4_F32. Wave-uniform guard
// keeps EXEC all-ones at every WMMA.

template<int K>
__global__ __launch_bounds__(256) void gemm_wmma(const float* __restrict__ A,
                                                 const float* __restrict__ Bt,
                                                 float* __restrict__ C, int M) {
  constexpr int KP = K + 4;  // pad LDS row stride -> conflict-free b64 reads
  __shared__ float lds[HID * KP];
  for (int idx = threadIdx.x; idx < HID * K; idx += 256) {
    int n = idx / K, k = idx % K;
    lds[n * KP + k] = Bt[idx];
  }
  __syncthreads();

  const int lane = threadIdx.x & 31;
  const int wave = threadIdx.x >> 5;
  const int half = lane >> 4;   // K-pair / row-half selector
  const int mn   = lane & 15;   // row (A) / col (B,C)
  const int tile = blockIdx.x * 8 + wave;
  if (tile * 16 >= M) return;   // uniform across the wave

  v8f acc0 = {}, acc1 = {}, acc2 = {}, acc3 = {};
  // A 16x4 f32 layout: lane = half*16+m holds A[m][2*half], A[m][2*half+1]
  const float* arow = A + (size_t)(tile * 16 + mn) * K + 2 * half;
  // B 4x16 layout (column-striped): lane = half*16+n holds B[2*half..+1][n]
  const float* brow = lds + mn * KP + 2 * half;

#pragma unroll 4
  for (int ks = 0; ks < K / 4; ++ks) {
    v2f a  = *(const v2f*)(arow + ks * 4);
    v2f b0 = *(const v2f*)(brow + 0 * 16 * KP + ks * 4);
    v2f b1 = *(const v2f*)(brow + 1 * 16 * KP + ks * 4);
    v2f b2 = *(const v2f*)(brow + 2 * 16 * KP + ks * 4);
    v2f b3 = *(const v2f*)(brow + 3 * 16 * KP + ks * 4);
    acc0 = __builtin_amdgcn_wmma_f32_16x16x4_f32(false, a, false, b0, (short)0, acc0, false, false);
    acc1 = __builtin_amdgcn_wmma_f32_16x16x4_f32(false, a, false, b1, (short)0, acc1, false, false);
    acc2 = __builtin_amdgcn_wmma_f32_16x16x4_f32(false, a, false, b2, (short)0, acc2, false, false);
    acc3 = __builtin_amdgcn_wmma_f32_16x16x4_f32(false, a, false, b3, (short)0, acc3, false, false);
  }

  // C/D layout: lanes 0-15 -> rows v, col=lane; lanes 16-31 -> rows v+8, col=lane-16
  float* crow = C + (size_t)(tile * 16 + 8 * half) * HID + mn;
#pragma unroll
  for (int v = 0; v < 8; ++v) {
    crow[(size_t)v * HID +  0] = acc0[v];
    crow[(size_t)v * HID + 16] = acc1[v];
    crow[(size_t)v * HID + 32] = acc2[v];
    crow[(size_t)v * HID + 48] = acc3[v];
  }
}

// ---------------- graph scatter (edge aggregation) ----------------
// thread per (edge, feature); wave covers 32 consecutive features of one edge.

__global__ void k_scatter(const long long* __restrict__ src,
                          const long long* __restrict__ dst,
                          const float* __restrict__ dinv,
                          const float* __restrict__ h,
                          float* __restrict__ agg, int total) {
  int idx = blockIdx.x * blockDim.x + threadIdx.x;
  if (idx >= total) return;
  int e = idx >> 6;
  int f = idx & 63;
  int s = (int)src[e];
  int d = (int)dst[e];
  float nrm = dinv[s] * dinv[d];
  atomicAdd(&agg[(size_t)d * HID + f], nrm * h[(size_t)s * HID + f]);
}

// h = relu(agg + dinv^2 * h0 + b1)   (self-loop term folded in), in-place on agg
__global__ void k_bias_relu(float* __restrict__ agg,
                            const float* __restrict__ h0,
                            const float* __restrict__ dinv,
                            const float* __restrict__ b1, int total) {
  int idx = blockIdx.x * blockDim.x + threadIdx.x;
  if (idx >= total) return;
  int i = idx >> 6;
  int f = idx & 63;
  float di = dinv[i];
  float v = agg[idx] + di * di * h0[idx] + b1[f];
  agg[idx] = v > 0.0f ? v : 0.0f;
}

// scatter second layer directly into d_out = [mu (N*32) | logstd (N*32)]
__global__ void k_scatter_out(const long long* __restrict__ src,
                              const long long* __restrict__ dst,
                              const float* __restrict__ dinv,
                              const float* __restrict__ h1,
                              float* __restrict__ out, int total, int N) {
  int idx = blockIdx.x * blockDim.x + threadIdx.x;
  if (idx >= total) return;
  int e = idx >> 6;
  int f = idx & 63;
  int s = (int)src[e];
  int d = (int)dst[e];
  float v = dinv[s] * dinv[d] * h1[(size_t)s * HID + f];
  if (f < C_OUT)
    atomicAdd(&out[(size_t)d * C_OUT + f], v);
  else
    atomicAdd(&out[(size_t)N * C_OUT + (size_t)d * C_OUT + (f - C_OUT)], v);
}

// add self-loop term and biases to both output halves
__global__ void k_finalize(float* __restrict__ out,
                           const float* __restrict__ h1,
                           const float* __restrict__ dinv,
                           const float* __restrict__ bmu,
                           const float* __restrict__ bls, int total, int N) {
  int idx = blockIdx.x * blockDim.x + threadIdx.x;
  if (idx >= total) return;
  int i = idx >> 6;
  int f = idx & 63;
  float di = dinv[i];
  float v = di * di * h1[idx];
  if (f < C_OUT)
    out[(size_t)i * C_OUT + f] += v + bmu[f];
  else
    out[(size_t)N * C_OUT + (size_t)i * C_OUT + (f - C_OUT)] += v + bls[f - C_OUT];
}

// ---------------- launch ----------------

extern "C" void kernel_launch(void* const* d_in, const int* in_sizes, int n_in,
                              void* d_out, int out_size, void* d_ws, size_t ws_size,
                              hipStream_t stream) {
  const float*     x   = (const float*)d_in[0];
  const long long* ei  = (const long long*)d_in[1];
  const float*     W1  = (const float*)d_in[2];
  const float*     b1  = (const float*)d_in[3];
  const float*     Wmu = (const float*)d_in[4];
  const float*     bmu = (const float*)d_in[5];
  const float*     Wls = (const float*)d_in[6];
  const float*     bls = (const float*)d_in[7];
  float* out = (float*)d_out;

  const int N = in_sizes[0] / C_IN;   // 50000
  const int E = in_sizes[1] / 2;      // 800000
  const long long* src = ei;
  const long long* dst = ei + E;

  // workspace layout (floats): dinv | Wt1 | Wt2 | h0/h1 | agg/h
  float* ws   = (float*)d_ws;
  float* dinv = ws;
  float* wt1  = dinv + ((N + 255) & ~255);
  float* wt2  = wt1 + C_IN * HID;
  float* h0   = wt2 + HID * HID;
  float* agg  = h0 + (size_t)N * HID;

  const int B = 256;
  const int totN = N * HID;   // 3.2M
  const int totE = E * HID;   // 51.2M

  // symmetric normalization: deg = indeg + 1 (self loop), dinv = rsqrt(deg)
  k_set1   <<<(N + B - 1) / B, B, 0, stream>>>(dinv, N);
  k_deg_acc<<<(E + B - 1) / B, B, 0, stream>>>(dst, dinv, E);
  k_rsqrt  <<<(N + B - 1) / B, B, 0, stream>>>(dinv, N);

  // pre-transpose weights for LDS-resident B operands
  k_transpose_w1<<<(C_IN * HID + B - 1) / B, B, 0, stream>>>(W1, wt1);
  k_build_wcat  <<<(HID * HID + B - 1) / B, B, 0, stream>>>(Wmu, Wls, wt2);

  const int tiles   = (N + 15) / 16;
  const int gblocks = (tiles + 7) / 8;

  // layer 1: h0 = x @ W1 (WMMA), then aggregate + bias + relu (in-place -> agg)
  gemm_wmma<C_IN><<<gblocks, 256, 0, stream>>>(x, wt1, h0, N);
  k_zero     <<<(totN + B - 1) / B, B, 0, stream>>>(agg, totN);
  k_scatter  <<<(totE + B - 1) / B, B, 0, stream>>>(src, dst, dinv, h0, agg, totE);
  k_bias_relu<<<(totN + B - 1) / B, B, 0, stream>>>(agg, h0, dinv, b1, totN);

  // layer 2: h1 = h @ [W_mu|W_ls] (WMMA, overwrites h0), aggregate into d_out
  gemm_wmma<HID><<<gblocks, 256, 0, stream>>>(agg, wt2, h0, N);
  k_zero       <<<(totN + B - 1) / B, B, 0, stream>>>(out, totN);
  k_scatter_out<<<(totE + B - 1) / B, B, 0, stream>>>(src, dst, dinv, h0, out, totE, N);
  k_finalize   <<<(totN + B - 1) / B, B, 0, stream>>>(out, h0, dinv, bmu, bls, totN, N);
}